// LearnedLSHAttention_68015102100110
// MI455X (gfx1250) — compile-verified
//
#include <hip/hip_runtime.h>
#include <hip/hip_bf16.h>

// ---------------- problem constants (match reference) ----------------
#define NN        20000
#define HIDDEN    128
#define HEADS     8
#define HEAD      16
#define LAP       16
#define NBUCKETS  1024
#define NBINS     32
#define MAXDEG    64
#define DIN       (HIDDEN + LAP)     // 144

#define P_MAX     (4u << 20)         // pair capacity (4M)
#define HASH_LOG2 22
#define HASH_SIZE (1u << HASH_LOG2)  // 4M slots, u64 each
#define HASH_EMPTY 0xFFFFFFFFFFFFFFFFull

typedef __attribute__((ext_vector_type(2))) float v2f;
typedef __attribute__((ext_vector_type(8))) float v8f;

// =====================================================================
// fp32 WMMA GEMM:  C(M,Nn) = act(A(M,K) @ B(K,Nn) + bias)
// One wave computes a 32x32 tile as 2x2 register-tiled 16x16 WMMAs
// (V_WMMA_F32_16X16X4_F32). Fragment layouts per CDNA5 ISA:
//   A frag: lane holds A[row16 + lane%16][kgrp + {0,1}], kgrp = 2*(lane/16)
//   B frag: lane holds B[kgrp + {0,1}][col16 + lane%16]
//   C/D:    component v at lane -> C[v + 8*(lane/16)][lane%16]
// Requires M % 32 == 0 and Nn % 32 == 0 and K % 4 == 0 (true for all uses).
// =====================================================================
__global__ void gemm_wmma_f32(const float* __restrict__ A,
                              const float* __restrict__ B,
                              const float* __restrict__ bias,
                              float* __restrict__ C,
                              int M, int K, int Nn, int relu)
{
    const int wave = (blockIdx.x * blockDim.x + threadIdx.x) >> 5;
    const int lane = threadIdx.x & 31;
    const int ntn  = Nn >> 5;                  // 32-wide N tiles
    const int tiles = (M >> 5) * ntn;
    if (wave >= tiles) return;                 // per-wave uniform -> EXEC all 1s inside

    const int tm = wave / ntn;
    const int tn = wave % ntn;

    const int l16  = lane & 15;
    const int kgrp = (lane >> 4) << 1;         // 0 or 2
    const int r0   = (tm << 5) + l16;          // A rows (i=0), +16 for i=1
    const int c0   = (tn << 5) + l16;          // B cols (j=0), +16 for j=1

    v8f acc00 = {}, acc01 = {}, acc10 = {}, acc11 = {};

    for (int k0 = 0; k0 < K; k0 += 4) {
        const int ka = k0 + kgrp;
        v2f a0, a1, b0, b1;
        a0.x = A[(size_t)r0 * K + ka];
        a0.y = A[(size_t)r0 * K + ka + 1];
        a1.x = A[(size_t)(r0 + 16) * K + ka];
        a1.y = A[(size_t)(r0 + 16) * K + ka + 1];
        b0.x = B[(size_t)ka * Nn + c0];
        b0.y = B[(size_t)(ka + 1) * Nn + c0];
        b1.x = B[(size_t)ka * Nn + c0 + 16];
        b1.y = B[(size_t)(ka + 1) * Nn + c0 + 16];
        acc00 = __builtin_amdgcn_wmma_f32_16x16x4_f32(false, a0, false, b0, (short)0, acc00, false, false);
        acc01 = __builtin_amdgcn_wmma_f32_16x16x4_f32(false, a0, false, b1, (short)0, acc01, false, false);
        acc10 = __builtin_amdgcn_wmma_f32_16x16x4_f32(false, a1, false, b0, (short)0, acc10, false, false);
        acc11 = __builtin_amdgcn_wmma_f32_16x16x4_f32(false, a1, false, b1, (short)0, acc11, false, false);
    }

    const int half = lane >> 4;
    const float bias0 = bias[c0];
    const float bias1 = bias[c0 + 16];
#pragma unroll
    for (int v = 0; v < 8; ++v) {
        const int m0 = (tm << 5) + v + (half << 3);       // i=0 rows
        const int m1 = m0 + 16;                           // i=1 rows
        float v00 = acc00[v] + bias0;
        float v01 = acc01[v] + bias1;
        float v10 = acc10[v] + bias0;
        float v11 = acc11[v] + bias1;
        if (relu) {
            v00 = fmaxf(v00, 0.0f); v01 = fmaxf(v01, 0.0f);
            v10 = fmaxf(v10, 0.0f); v11 = fmaxf(v11, 0.0f);
        }
        C[(size_t)m0 * Nn + c0]      = v00;
        C[(size_t)m0 * Nn + c0 + 16] = v01;
        C[(size_t)m1 * Nn + c0]      = v10;
        C[(size_t)m1 * Nn + c0 + 16] = v11;
    }
}

// ---------------- utility fills ----------------
__global__ void fill_u32(unsigned* p, unsigned v, long long n) {
    long long i = (long long)blockIdx.x * blockDim.x + threadIdx.x;
    if (i < n) p[i] = v;
}
__global__ void fill_u64(unsigned long long* p, unsigned long long v, long long n) {
    long long i = (long long)blockIdx.x * blockDim.x + threadIdx.x;
    if (i < n) p[i] = v;
}

// ---------------- concat([X, lap]) ----------------
__global__ void pack_concat(const float* __restrict__ X,
                            const float* __restrict__ lap,
                            float* __restrict__ dst, int n)
{
    int i = blockIdx.x * blockDim.x + threadIdx.x;
    if (i >= n * DIN) return;
    int r = i / DIN, c = i % DIN;
    dst[i] = (c < HIDDEN) ? X[r * HIDDEN + c] : lap[r * LAP + (c - HIDDEN)];
}

// ---------------- row argmax (1024 cols), one wave per row ----------------
__global__ void argmax_rows(const float* __restrict__ L, int* __restrict__ out,
                            int n, int cols)
{
    int wave = (blockIdx.x * blockDim.x + threadIdx.x) >> 5;
    int lane = threadIdx.x & 31;
    if (wave >= n) return;
    const float* row = L + (size_t)wave * cols;
    float best = -__builtin_inff();
    int bi = 0x7FFFFFFF;
    for (int c = lane; c < cols; c += 32) {
        float v = row[c];
        if (v > best) { best = v; bi = c; }
    }
#pragma unroll
    for (int o = 16; o > 0; o >>= 1) {
        float ov = __shfl_down(best, o, 32);
        int   oi = __shfl_down(bi,   o, 32);
        if (ov > best || (ov == best && oi < bi)) { best = ov; bi = oi; }
    }
    if (lane == 0) out[wave] = bi;
}

// ---------------- counting sort of nodes into buckets ----------------
__global__ void bucket_count(const int* __restrict__ b, int* cnt, int n) {
    int i = blockIdx.x * blockDim.x + threadIdx.x;
    if (i < n) atomicAdd(&cnt[b[i]], 1);
}

__global__ void scan1024(const int* __restrict__ cnt, int* __restrict__ off) {
    __shared__ int tmp[NBUCKETS];
    int t = threadIdx.x;
    tmp[t] = cnt[t];
    __syncthreads();
    for (int d = 1; d < NBUCKETS; d <<= 1) {
        int v = (t >= d) ? tmp[t - d] : 0;
        __syncthreads();
        tmp[t] += v;
        __syncthreads();
    }
    off[t + 1] = tmp[t];
    if (t == 0) off[0] = 0;
}

__global__ void scatter_nodes(const int* __restrict__ b, const int* __restrict__ off,
                              int* cursor, int* __restrict__ list, int n) {
    int i = blockIdx.x * blockDim.x + threadIdx.x;
    if (i >= n) return;
    int bk = b[i];
    list[off[bk] + atomicAdd(&cursor[bk], 1)] = i;
}

// ---------------- dedup hash-set insert (pair code = s*N + d) ----------------
__device__ __forceinline__ void pair_insert(unsigned long long* table,
                                            int* psrc, int* pdst, int* pcount,
                                            int s, int d)
{
    unsigned long long code = (unsigned long long)s * (unsigned long long)NN +
                              (unsigned long long)d;
    unsigned slot = (unsigned)((code * 0x9E3779B97F4A7C15ull) >> 40) & (HASH_SIZE - 1);
    for (int probe = 0; probe < 4096; ++probe) {
        unsigned long long prev = atomicCAS(&table[slot], HASH_EMPTY, code);
        if (prev == HASH_EMPTY) {
            unsigned i = atomicAdd((unsigned*)pcount, 1u);
            if (i < P_MAX) { psrc[i] = s; pdst[i] = d; }
            return;
        }
        if (prev == code) return;
        slot = (slot + 1) & (HASH_SIZE - 1);
    }
}

__global__ void edge_pairs(const int* __restrict__ ei, int E,
                           unsigned long long* table, int* psrc, int* pdst, int* pcount)
{
    int e = blockIdx.x * blockDim.x + threadIdx.x;
    if (e >= E) return;
    pair_insert(table, psrc, pdst, pcount, ei[e], ei[E + e]);
}

__global__ void lsh_pairs(const int* __restrict__ qoff, const int* __restrict__ qlist,
                          const int* __restrict__ koff, const int* __restrict__ klist,
                          unsigned long long* table, int* psrc, int* pdst, int* pcount)
{
    int b = blockIdx.x;
    int qs = qoff[b], nq = qoff[b + 1] - qs;
    int ks = koff[b], nk = koff[b + 1] - ks;
    long long total = (long long)nq * (long long)nk;
    for (long long t = threadIdx.x; t < total; t += blockDim.x) {
        int qi = qlist[qs + (int)(t / nk)];
        int kj = klist[ks + (int)(t % nk)];
        if (qi == kj) continue;
        pair_insert(table, psrc, pdst, pcount, qi, kj);
    }
}

// ---------------- score recompute ----------------
__device__ __forceinline__ float pair_score(
    int p, int h,
    const float* __restrict__ Q, const float* __restrict__ K,
    const float* __restrict__ lap, const float* __restrict__ spd_bias,
    const float* __restrict__ dse, const float* __restrict__ dde,
    const float* __restrict__ bounds, const int* __restrict__ deg,
    const int* __restrict__ psrc, const int* __restrict__ pdst)
{
    int s = psrc[p], d = pdst[p];
    float dot = 0.0f;
#pragma unroll
    for (int t = 0; t < HEAD; ++t)
        dot += Q[(size_t)s * HIDDEN + h * HEAD + t] * K[(size_t)d * HIDDEN + h * HEAD + t];
    float sc = dot * 0.25f;                       // 1/sqrt(16)

    float d2 = 0.0f;
#pragma unroll
    for (int t = 0; t < LAP; ++t) {
        float df = lap[s * LAP + t] - lap[d * LAP + t];
        d2 += df * df;
    }
    float dist = sqrtf(d2);
    int idx = 0;
#pragma unroll
    for (int t = 0; t < NBINS; ++t) idx += (bounds[t] < dist) ? 1 : 0;  // searchsorted left
    sc += spd_bias[idx * HEADS + h];

    int ds = min(max(deg[s], 0), MAXDEG + 1);
    int dd = min(max(deg[d], 0), MAXDEG + 1);
    sc += dse[ds * HEADS + h] + dde[dd * HEADS + h];
    return sc;
}

__device__ __forceinline__ unsigned fmap(float f) {
    unsigned u = __float_as_uint(f);
    return (u & 0x80000000u) ? ~u : (u | 0x80000000u);
}

#define SCORE_ARGS Q, K, lap, spd_bias, dse, dde, bounds, deg, psrc, pdst
#define SCORE_PARAMS const float* __restrict__ Q, const float* __restrict__ K,            \
    const float* __restrict__ lap, const float* __restrict__ spd_bias,                    \
    const float* __restrict__ dse, const float* __restrict__ dde,                         \
    const float* __restrict__ bounds, const int* __restrict__ deg,                        \
    const int* __restrict__ psrc, const int* __restrict__ pdst, const int* __restrict__ pcount

__global__ void attn_max(SCORE_PARAMS, unsigned* __restrict__ m_u)
{
    long long idx = (long long)blockIdx.x * blockDim.x + threadIdx.x;
    long long P = min(*(const unsigned*)pcount, P_MAX);
    if (idx >= P * HEADS) return;
    int p = (int)(idx >> 3), h = (int)(idx & 7);
    float sc = pair_score(p, h, SCORE_ARGS);
    atomicMax(&m_u[psrc[p] * HEADS + h], fmap(sc));
}

__global__ void finalize_m(unsigned* __restrict__ m_u, int n)
{
    int i = blockIdx.x * blockDim.x + threadIdx.x;
    if (i >= n) return;
    unsigned u = m_u[i];
    float f;
    if (u == 0u) f = 0.0f;
    else if (u & 0x80000000u) f = __uint_as_float(u ^ 0x80000000u);
    else f = __uint_as_float(~u);
    if (isinf(f) && f < 0.0f) f = 0.0f;           // where(isneginf(m), 0, m)
    m_u[i] = __float_as_uint(f);                  // in-place: now holds float m
}

__global__ void attn_denom(SCORE_PARAMS, const float* __restrict__ m,
                           float* __restrict__ denom)
{
    long long idx = (long long)blockIdx.x * blockDim.x + threadIdx.x;
    long long P = min(*(const unsigned*)pcount, P_MAX);
    if (idx >= P * HEADS) return;
    int p = (int)(idx >> 3), h = (int)(idx & 7);
    float sc = pair_score(p, h, SCORE_ARGS);
    atomicAdd(&denom[psrc[p] * HEADS + h], expf(sc - m[psrc[p] * HEADS + h]));
}

__global__ void attn_accum(SCORE_PARAMS, const float* __restrict__ m,
                           const float* __restrict__ denom,
                           const float* __restrict__ V, float* __restrict__ hacc)
{
    long long idx = (long long)blockIdx.x * blockDim.x + threadIdx.x;
    long long P = min(*(const unsigned*)pcount, P_MAX);
    if (idx >= P * HEADS) return;
    int p = (int)(idx >> 3), h = (int)(idx & 7);
    int s = psrc[p], d = pdst[p];
    float sc = pair_score(p, h, SCORE_ARGS);
    float w = expf(sc - m[s * HEADS + h]) / (denom[s * HEADS + h] + 1e-16f);
#pragma unroll
    for (int t = 0; t < HEAD; ++t)
        atomicAdd(&hacc[(size_t)s * HIDDEN + h * HEAD + t],
                  w * V[(size_t)d * HIDDEN + h * HEAD + t]);
}

// =====================================================================
extern "C" void kernel_launch(void* const* d_in, const int* in_sizes, int n_in,
                              void* d_out, int out_size, void* d_ws, size_t ws_size,
                              hipStream_t stream)
{
    const float* Q      = (const float*)d_in[0];
    const float* K      = (const float*)d_in[1];
    const float* V      = (const float*)d_in[2];
    const float* lap    = (const float*)d_in[3];
    const float* W_q1   = (const float*)d_in[4];
    const float* b_q1   = (const float*)d_in[5];
    const float* W_q2   = (const float*)d_in[6];
    const float* b_q2   = (const float*)d_in[7];
    const float* W_k1   = (const float*)d_in[8];
    const float* b_k1   = (const float*)d_in[9];
    const float* W_k2   = (const float*)d_in[10];
    const float* b_k2   = (const float*)d_in[11];
    const float* spd_b  = (const float*)d_in[12];
    const float* dse    = (const float*)d_in[13];
    const float* dde    = (const float*)d_in[14];
    const float* W_out  = (const float*)d_in[15];
    const float* b_out  = (const float*)d_in[16];
    const float* bounds = (const float*)d_in[17];
    const int*   eidx   = (const int*)d_in[18];
    const int*   deg    = (const int*)d_in[19];
    const int E = in_sizes[18] / 2;

    float* h_global = (float*)d_out;                              // (N,128)
    float* l_q      = h_global + (size_t)NN * HIDDEN;             // (N,1024)
    float* l_k      = l_q + (size_t)NN * NBUCKETS;                // (N,1024)

    // -------- workspace carve-out --------
    char* w = (char*)d_ws;
    auto carve = [&](size_t bytes) {
        void* p = (void*)w;
        w += (bytes + 255) & ~(size_t)255;
        return p;
    };
    float* qin  = (float*)carve((size_t)NN * DIN * 4);
    float* kin  = (float*)carve((size_t)NN * DIN * 4);
    float* hq   = (float*)carve((size_t)NN * HIDDEN * 4);
    float* hk   = (float*)carve((size_t)NN * HIDDEN * 4);
    int*   bq   = (int*)carve((size_t)NN * 4);
    int*   bk   = (int*)carve((size_t)NN * 4);
    int*   qcnt = (int*)carve(NBUCKETS * 4);
    int*   kcnt = (int*)carve(NBUCKETS * 4);
    int*   qoff = (int*)carve((NBUCKETS + 1) * 4);
    int*   koff = (int*)carve((NBUCKETS + 1) * 4);
    int*   qcur = (int*)carve(NBUCKETS * 4);
    int*   kcur = (int*)carve(NBUCKETS * 4);
    int*   qlist = (int*)carve((size_t)NN * 4);
    int*   klist = (int*)carve((size_t)NN * 4);
    int*   pcount = (int*)carve(256);
    int*   psrc = (int*)carve((size_t)P_MAX * 4);
    int*   pdst = (int*)carve((size_t)P_MAX * 4);
    unsigned long long* htab = (unsigned long long*)carve((size_t)HASH_SIZE * 8);
    unsigned* m_u  = (unsigned*)carve((size_t)NN * HEADS * 4);
    float*   denom = (float*)carve((size_t)NN * HEADS * 4);
    float*   hacc  = (float*)carve((size_t)NN * HIDDEN * 4);

    const int T = 256;
    auto blks = [](long long n, int t) { return (unsigned)((n + t - 1) / t); };

    // -------- init scratch --------
    fill_u32<<<blks(NBUCKETS, T), T, 0, stream>>>((unsigned*)qcnt, 0, NBUCKETS);
    fill_u32<<<blks(NBUCKETS, T), T, 0, stream>>>((unsigned*)kcnt, 0, NBUCKETS);
    fill_u32<<<blks(NBUCKETS, T), T, 0, stream>>>((unsigned*)qcur, 0, NBUCKETS);
    fill_u32<<<blks(NBUCKETS, T), T, 0, stream>>>((unsigned*)kcur, 0, NBUCKETS);
    fill_u32<<<1, 1, 0, stream>>>((unsigned*)pcount, 0, 1);
    fill_u64<<<blks(HASH_SIZE, T), T, 0, stream>>>(htab, HASH_EMPTY, HASH_SIZE);
    fill_u32<<<blks((long long)NN * HEADS, T), T, 0, stream>>>(m_u, 0, (long long)NN * HEADS);
    fill_u32<<<blks((long long)NN * HEADS, T), T, 0, stream>>>((unsigned*)denom, 0, (long long)NN * HEADS);
    fill_u32<<<blks((long long)NN * HIDDEN, T), T, 0, stream>>>((unsigned*)hacc, 0, (long long)NN * HIDDEN);

    // -------- MLPs (WMMA GEMMs) --------
    pack_concat<<<blks((long long)NN * DIN, T), T, 0, stream>>>(Q, lap, qin, NN);
    pack_concat<<<blks((long long)NN * DIN, T), T, 0, stream>>>(K, lap, kin, NN);

    auto gemm = [&](const float* A, const float* B, const float* bias, float* C,
                    int M, int Kk, int Nn, int relu) {
        long long waves = (long long)(M / 32) * (Nn / 32);
        gemm_wmma_f32<<<blks(waves * 32, T), T, 0, stream>>>(A, B, bias, C, M, Kk, Nn, relu);
    };
    gemm(qin, W_q1, b_q1, hq, NN, DIN, HIDDEN, 1);
    gemm(kin, W_k1, b_k1, hk, NN, DIN, HIDDEN, 1);
    gemm(hq, W_q2, b_q2, l_q, NN, HIDDEN, NBUCKETS, 0);
    gemm(hk, W_k2, b_k2, l_k, NN, HIDDEN, NBUCKETS, 0);

    // -------- buckets --------
    argmax_rows<<<blks((long long)NN * 32, T), T, 0, stream>>>(l_q, bq, NN, NBUCKETS);
    argmax_rows<<<blks((long long)NN * 32, T), T, 0, stream>>>(l_k, bk, NN, NBUCKETS);

    bucket_count<<<blks(NN, T), T, 0, stream>>>(bq, qcnt, NN);
    bucket_count<<<blks(NN, T), T, 0, stream>>>(bk, kcnt, NN);
    scan1024<<<1, NBUCKETS, 0, stream>>>(qcnt, qoff);
    scan1024<<<1, NBUCKETS, 0, stream>>>(kcnt, koff);
    scatter_nodes<<<blks(NN, T), T, 0, stream>>>(bq, qoff, qcur, qlist, NN);
    scatter_nodes<<<blks(NN, T), T, 0, stream>>>(bk, koff, kcur, klist, NN);

    // -------- pair building (edges + LSH, deduped via hash set) --------
    edge_pairs<<<blks(E, T), T, 0, stream>>>(eidx, E, htab, psrc, pdst, pcount);
    lsh_pairs<<<NBUCKETS, T, 0, stream>>>(qoff, qlist, koff, klist, htab, psrc, pdst, pcount);

    // -------- segment softmax attention (3 recompute passes) --------
    long long maxwork = (long long)P_MAX * HEADS;
    attn_max<<<blks(maxwork, T), T, 0, stream>>>(Q, K, lap, spd_b, dse, dde, bounds, deg,
                                                 psrc, pdst, pcount, m_u);
    finalize_m<<<blks((long long)NN * HEADS, T), T, 0, stream>>>(m_u, NN * HEADS);
    attn_denom<<<blks(maxwork, T), T, 0, stream>>>(Q, K, lap, spd_b, dse, dde, bounds, deg,
                                                   psrc, pdst, pcount, (const float*)m_u, denom);
    attn_accum<<<blks(maxwork, T), T, 0, stream>>>(Q, K, lap, spd_b, dse, dde, bounds, deg,
                                                   psrc, pdst, pcount, (const float*)m_u, denom,
                                                   V, hacc);

    // -------- output projection (WMMA) --------
    gemm(hacc, W_out, b_out, h_global, NN, HIDDEN, HIDDEN, 0);
}